// RoIAlign_31421980737572
// MI455X (gfx1250) — compile-verified
//
#include <hip/hip_runtime.h>

typedef __attribute__((ext_vector_type(2))) float v2f;
typedef __attribute__((ext_vector_type(8))) float v8f;

#define OUT_HW   7
#define NBINS    49
#define SAMP     2
#define NPTS     14          // OUT_HW * SAMP per axis
#define NPOINTS  196         // 14 * 14 sample points per roi
#define CDIM     256
#define HDIM     200
#define WDIM     200
#define RSCALE   0.125f
#define CCHUNK   128         // channels per pass (2 passes)
#define CPAD     136         // row pitch: 2*CPAD % 64 == 16 -> conflict-free B loads
#define NPASS    (CDIM / CCHUNK)

__global__ __launch_bounds__(256) void roi_align_wmma(
    const float* __restrict__ feat,   // [N, C, H, W]
    const float* __restrict__ rois,   // [K, 5] = b, x1, y1, x2, y2
    float* __restrict__ out)          // [K, C, 7, 7]
{
  extern __shared__ float sval[];     // [NPOINTS][CPAD], bin-major point order
  __shared__ int   s_lo[2][NPTS];     // [y/x][sample]
  __shared__ int   s_hi[2][NPTS];
  __shared__ float s_fr[2][NPTS];     // fractional part
  __shared__ int   s_va[2][NPTS];     // valid flag
  __shared__ int   s_b;

  const int k = blockIdx.x;
  const int t = threadIdx.x;

  // ---- per-roi geometry, computed once (threads 0..27) ----
  if (t < 2 * NPTS) {
    const int axis = t / NPTS;        // 0 = y, 1 = x
    const int s    = t - axis * NPTS;
    const float* r = rois + (size_t)k * 5;
    if (t == 0) s_b = (int)r[0];
    const float a1 = (axis == 0 ? r[2] : r[1]) * RSCALE;
    const float a2 = (axis == 0 ? r[4] : r[3]) * RSCALE;
    const int   sz = (axis == 0 ? HDIM : WDIM);
    const float roi_sz = fmaxf(a2 - a1, 1.0f);   // aligned=False min size 1
    const float bin_sz = roi_sz / (float)OUT_HW;
    const int   pbin = s >> 1;
    const float sub  = (float)(s & 1) + 0.5f;
    float coord = a1 + (float)pbin * bin_sz + sub * bin_sz / (float)SAMP;
    const int valid = (coord >= -1.0f) && (coord <= (float)sz);
    float c = fmaxf(coord, 0.0f);
    int lo = (int)floorf(c);
    int hi;
    if (lo >= sz - 1) { lo = sz - 1; hi = sz - 1; c = (float)(sz - 1); }
    else              { hi = lo + 1; }
    s_lo[axis][s] = lo;
    s_hi[axis][s] = hi;
    s_fr[axis][s] = c - (float)lo;
    s_va[axis][s] = valid;
  }
  __syncthreads();

  const int lane   = t & 31;
  const int wave_u = __builtin_amdgcn_readfirstlane(t >> 5);  // uniform wave id 0..7
  const int c_loc  = t & (CCHUNK - 1);        // channel within chunk (phase 1)
  const int p0     = (t >> 7) * (NPOINTS / 2);// waves 0-3: pts 0..97, 4-7: 98..195
  const int b      = s_b;

  for (int pass = 0; pass < NPASS; ++pass) {
    // -------- phase 1: bilinear samples for 128 channels -> LDS --------
    {
      const int cg = pass * CCHUNK + c_loc;
      const float* fb = feat + ((size_t)(b * CDIM + cg)) * (HDIM * WDIM);
      for (int pp = 0; pp < NPOINTS / 2; ++pp) {
        const int p   = p0 + pp;           // bin-major point id: p = bin*4 + q
        const int bin = p >> 2, q = p & 3;
        const int ph  = bin / OUT_HW, pw = bin - ph * OUT_HW;
        const int py  = 2 * ph + (q >> 1);
        const int px  = 2 * pw + (q & 1);
        const int   ylo = s_lo[0][py], yhi = s_hi[0][py];
        const float ly  = s_fr[0][py];
        const int   vy  = s_va[0][py];
        const int   xlo = s_lo[1][px], xhi = s_hi[1][px];
        const float lx  = s_fr[1][px];
        const int   vx  = s_va[1][px];
        const float hy = 1.0f - ly, hx = 1.0f - lx;
        const float f00 = fb[ylo * WDIM + xlo];
        const float f01 = fb[ylo * WDIM + xhi];
        const float f10 = fb[yhi * WDIM + xlo];
        const float f11 = fb[yhi * WDIM + xhi];
        float v = hy * hx * f00 + hy * lx * f01 + ly * hx * f10 + ly * lx * f11;
        v = (vy && vx) ? v : 0.0f;
        sval[p * CPAD + c_loc] = v;
      }
    }
    __syncthreads();

    // -------- phase 2: 2x2-sample average pool as WMMA  D = P x val' --------
    // bins 0..47 = 3 full 16-bin M-tiles x 8 channel N-tiles = 24 tiles/pass.
    // Uniform bounds -> full EXEC around every v_wmma, fully unrolled k-loop.
    for (int tile = wave_u; tile < 24; tile += 8) {          // 3 tiles per wave
      const int mt = tile >> 3;            // 0..2 (scalar)
      const int nt = tile & 7;             // 0..7 (scalar)
      v8f acc = {};
      const int col    = nt * 16 + (lane & 15);   // B column = channel in chunk
      const int rowsel = (lane >> 4) << 1;        // lanes>=16 hold B rows k=2,3
#pragma unroll
      for (int ks = 0; ks < 16; ++ks) {
        // A[m,k] = 0.25 iff m == ks (all k): same value in both A VGPRs
        const float a = ((lane & 15) == ks) ? 0.25f : 0.0f;
        v2f A; A.x = a; A.y = a;
        // B[k,n] = val'[bin*4 + k][chan n], bin = 16*mt + ks  (rows <= 191)
        const int rb = (mt * 16 + ks) * 4 + rowsel;
        v2f B;
        B.x = sval[rb * CPAD + col];
        B.y = sval[(rb + 1) * CPAD + col];
        acc = __builtin_amdgcn_wmma_f32_16x16x4_f32(false, A, false, B,
                                                    (short)0, acc, false, false);
      }
      // D layout: vgpr r, lane L -> (m = r + 8*(L>=16), n = L&15).
      // All bins here are < 48: unconditional stores, no exec juggling.
      const int cg = pass * CCHUNK + nt * 16 + (lane & 15);
      float* ob = out + ((size_t)k * CDIM + cg) * NBINS + (mt * 16 + 8 * (lane >> 4));
#pragma unroll
      for (int r = 0; r < 8; ++r) ob[r] = acc[r];
    }

    // -------- tail: bin 48 (rows 192..195) via plain VALU --------
    if (t < CCHUNK) {
      const float v = 0.25f * (sval[192 * CPAD + t] + sval[193 * CPAD + t] +
                               sval[194 * CPAD + t] + sval[195 * CPAD + t]);
      const int cg = pass * CCHUNK + t;
      out[((size_t)k * CDIM + cg) * NBINS + 48] = v;
    }
    __syncthreads();   // before next pass overwrites LDS
  }
}

extern "C" void kernel_launch(void* const* d_in, const int* in_sizes, int n_in,
                              void* d_out, int out_size, void* d_ws, size_t ws_size,
                              hipStream_t stream) {
  const float* feat = (const float*)d_in[0];
  const float* rois = (const float*)d_in[1];
  float* out = (float*)d_out;
  const int K = in_sizes[1] / 5;                             // 1000
  const size_t lds = (size_t)NPOINTS * CPAD * sizeof(float); // ~104 KB dynamic LDS
  roi_align_wmma<<<K, 256, lds, stream>>>(feat, rois, out);
}